// ElectrostaticDensity_77335181132475
// MI455X (gfx1250) — compile-verified
//
#include <hip/hip_runtime.h>

#define GRID_M   256
#define NBATCH   8
#define GRID2    (GRID_M * GRID_M)   // 65536

typedef __attribute__((ext_vector_type(2))) float v2f;
typedef __attribute__((ext_vector_type(8))) float v8f;

// Normalized 1-D Gaussian weight for sigma=2, ksize=13 (band |d|<=6).
// w[d] = exp(-d^2/8) / 5.0081226; exp(-d^2/8) = exp2(-0.18033688*d^2)
__device__ __forceinline__ float gw(int d) {
    int d2 = d * d;
    float v = 0.19967565f * exp2f(-0.18033688f * (float)d2);
    return (d2 <= 36) ? v : 0.0f;
}

__device__ __forceinline__ void gatomicAdd(float* p, float v) {
    // native global_atomic_add_f32 (no CAS loop)
    unsafeAtomicAdd(p, v);
}

// ---------------------------------------------------------------- zero ws
__global__ void __launch_bounds__(256)
zero_kernel(float* __restrict__ D, float* __restrict__ acc, int n) {
    int i = blockIdx.x * blockDim.x + threadIdx.x;
    if (i < n)  D[i]   = 0.0f;
    if (i < 16) acc[i] = 0.0f;
}

// ---------------------------------------------- scatter + boundary reduce
__global__ void __launch_bounds__(256)
scatter_kernel(const float* __restrict__ pos,
               const float* __restrict__ siz,
               const unsigned char* __restrict__ msk,
               float* __restrict__ Dgrid,
               float* __restrict__ bndAcc,
               int total, int N) {
    __shared__ float sb[NBATCH];
    const int tid = threadIdx.x;
    if (tid < NBATCH) sb[tid] = 0.0f;
    __syncthreads();

    const int idx = blockIdx.x * blockDim.x + tid;
    if (idx < total) {
        const int b = idx / N;
        const float2 p = ((const float2*)pos)[idx];
        const float2 s = ((const float2*)siz)[idx];
        const float keep = msk[idx] ? 0.0f : 1.0f;

        // density splat
        const float gx = (p.x + 1.0f) * 127.5f;      // *(M-1)/2
        const float gy = (p.y + 1.0f) * 127.5f;
        const float macro = s.x * s.y * 16384.0f * keep;  // / (2/M)^2
        int x0 = (int)floorf(gx); x0 = x0 < 0 ? 0 : (x0 > 255 ? 255 : x0);
        int y0 = (int)floorf(gy); y0 = y0 < 0 ? 0 : (y0 > 255 ? 255 : y0);
        const int x1 = x0 + 1 > 255 ? 255 : x0 + 1;
        const int y1 = y0 + 1 > 255 ? 255 : y0 + 1;
        const float wx = gx - (float)x0;
        const float wy = gy - (float)y0;
        float* Db = Dgrid + b * GRID2;
        gatomicAdd(&Db[y0 * GRID_M + x0], (1.0f - wx) * (1.0f - wy) * macro);
        gatomicAdd(&Db[y1 * GRID_M + x0], (1.0f - wx) * wy          * macro);
        gatomicAdd(&Db[y0 * GRID_M + x1], wx          * (1.0f - wy) * macro);
        gatomicAdd(&Db[y1 * GRID_M + x1], wx          * wy          * macro);

        // boundary violation^2 (rare: only points within half-size of the walls)
        const float hx = 0.5f * s.x, hy = 0.5f * s.y;
        const float bx = fmaxf(-1.0f - (p.x - hx), 0.0f) + fmaxf(p.x + hx - 1.0f, 0.0f);
        const float by = fmaxf(-1.0f - (p.y - hy), 0.0f) + fmaxf(p.y + hy - 1.0f, 0.0f);
        const float v  = (bx * bx + by * by) * keep;   // keep^2 == keep
        if (v != 0.0f) atomicAdd(&sb[b], v);           // LDS ds_add_f32
    }
    __syncthreads();
    if (tid < NBATCH) {
        const float v = sb[tid];
        if (v != 0.0f) gatomicAdd(&bndAcc[tid], v);
    }
}

// ---------------------------------------------------------------------------
// Separable Gaussian as banded GEMM on WMMA: S = G * D * G, G[i][j]=w[|i-j|].
// One wave computes one 16x16 tile. Fixed 8 K-blocks (32 rows) always cover
// the 28-wide band; gw()==0 outside the band kills out-of-band products, so
// the loop is branch-free, fully unrolled, and EXEC stays all-ones (WMMA req).
// ---------------------------------------------------------------------------

// pass 1: T = D * G   (smooth along x)
__global__ void __launch_bounds__(256)
conv_rows_wmma(const float* __restrict__ Din, float* __restrict__ Tout) {
    const int wave = __builtin_amdgcn_readfirstlane(
        (blockIdx.x * blockDim.x + threadIdx.x) >> 5);   // scalar tile id
    const int lane = threadIdx.x & 31;
    const int half = lane >> 4;
    const int lm   = lane & 15;
    const int b    = wave >> 8;
    const int tile = wave & 255;
    const int I    = tile >> 4;
    const int J    = tile & 15;

    const float* D = Din  + b * GRID2;
    float*       T = Tout + b * GRID2;

    const int colBase = J << 4;
    const int col  = colBase + lm;
    const int aRow = (I << 4) + lm;
    int kb0 = (colBase - 6) >> 2;                 // arithmetic shift ok
    kb0 = kb0 < 0 ? 0 : (kb0 > 56 ? 56 : kb0);    // fixed 8-block window

    v8f c = {0.f, 0.f, 0.f, 0.f, 0.f, 0.f, 0.f, 0.f};
#pragma unroll
    for (int u = 0; u < 8; ++u) {
        const int k = ((kb0 + u) << 2) + (half << 1);
        // A lanes: D[aRow][k], D[aRow][k+1]  (contiguous -> b64 load)
        v2f a = *(const v2f*)(D + aRow * GRID_M + k);
        // B lanes: G[k][col], G[k+1][col]  (generated in registers)
        v2f g;
        g.x = gw(col - k);
        g.y = gw(col - (k + 1));
        c = __builtin_amdgcn_wmma_f32_16x16x4_f32(false, a, false, g,
                                                  (short)0, c, false, false);
    }
#pragma unroll
    for (int i = 0; i < 8; ++i) {
        const int row = (I << 4) + i + (half << 3);
        T[row * GRID_M + col] = c[i];
    }
}

// pass 2: S = G * T   (smooth along y)
__global__ void __launch_bounds__(256)
conv_cols_wmma(const float* __restrict__ Tin, float* __restrict__ Sout) {
    const int wave = __builtin_amdgcn_readfirstlane(
        (blockIdx.x * blockDim.x + threadIdx.x) >> 5);
    const int lane = threadIdx.x & 31;
    const int half = lane >> 4;
    const int lm   = lane & 15;
    const int b    = wave >> 8;
    const int tile = wave & 255;
    const int I    = tile >> 4;
    const int J    = tile & 15;

    const float* T = Tin  + b * GRID2;
    float*       S = Sout + b * GRID2;

    const int rowBase = I << 4;
    const int aRow = rowBase + lm;
    const int col  = (J << 4) + lm;
    int kb0 = (rowBase - 6) >> 2;
    kb0 = kb0 < 0 ? 0 : (kb0 > 56 ? 56 : kb0);

    v8f c = {0.f, 0.f, 0.f, 0.f, 0.f, 0.f, 0.f, 0.f};
#pragma unroll
    for (int u = 0; u < 8; ++u) {
        const int k = ((kb0 + u) << 2) + (half << 1);
        // A lanes: G[aRow][k], G[aRow][k+1]  (generated in registers)
        v2f a;
        a.x = gw(aRow - k);
        a.y = gw(aRow - (k + 1));
        // B lanes: T[k][col], T[k+1][col]
        v2f g;
        g.x = T[k * GRID_M + col];
        g.y = T[(k + 1) * GRID_M + col];
        c = __builtin_amdgcn_wmma_f32_16x16x4_f32(false, a, false, g,
                                                  (short)0, c, false, false);
    }
#pragma unroll
    for (int i = 0; i < 8; ++i) {
        const int row = rowBase + i + (half << 3);
        S[row * GRID_M + col] = c[i];
    }
}

// --------------------------------------- per-batch sum of relu(S-1)^2
__global__ void __launch_bounds__(256)
overflow_kernel(const float* __restrict__ S, float* __restrict__ ovAcc) {
    __shared__ float sb[NBATCH];
    const int tid = threadIdx.x;
    if (tid < NBATCH) sb[tid] = 0.0f;
    __syncthreads();
    const int idx = blockIdx.x * blockDim.x + tid;
    const int b = idx >> 16;                 // 65536 cells per batch; wave-uniform
    const float r = fmaxf(S[idx] - 1.0f, 0.0f);
    float v = r * r;
#pragma unroll
    for (int o = 16; o > 0; o >>= 1)         // wave32 shuffle reduction
        v += __shfl_xor(v, o, 32);
    if ((tid & 31) == 0 && v != 0.0f) atomicAdd(&sb[b], v);
    __syncthreads();
    if (tid < NBATCH) {
        const float v2 = sb[tid];
        if (v2 != 0.0f) gatomicAdd(&ovAcc[tid], v2);
    }
}

// ----------------------------------------------------- potential = ov + 10*bnd
__global__ void __launch_bounds__(32)
finalize_kernel(const float* __restrict__ acc, float* __restrict__ out) {
    const int t = threadIdx.x;
    if (t < NBATCH) out[t] = acc[NBATCH + t] + 10.0f * acc[t];
}

extern "C" void kernel_launch(void* const* d_in, const int* in_sizes, int n_in,
                              void* d_out, int out_size, void* d_ws, size_t ws_size,
                              hipStream_t stream) {
    const float*         positions = (const float*)d_in[0];
    const float*         sizes     = (const float*)d_in[1];
    const unsigned char* mask      = (const unsigned char*)d_in[2];
    float* out = (float*)d_out;
    float* ws  = (float*)d_ws;

    const int totalPts = in_sizes[0] / 2;     // B*N = 4,000,000
    const int N        = totalPts / NBATCH;   // 500,000

    float* Draw   = ws;                        // raw density, 8*65536 f32
    float* Ttmp   = ws + NBATCH * GRID2;       // row-smoothed temp
    float* acc    = ws + 2 * NBATCH * GRID2;   // [0..7]=boundary, [8..15]=overflow
    float* Smooth = out + NBATCH;              // density_smooth output region

    const int nD = NBATCH * GRID2;             // 524288

    zero_kernel<<<(nD + 255) / 256, 256, 0, stream>>>(Draw, acc, nD);

    scatter_kernel<<<(totalPts + 255) / 256, 256, 0, stream>>>(
        positions, sizes, mask, Draw, acc, totalPts, N);

    // 2048 tiles, 1 wave each, 8 waves per block -> 256 blocks
    conv_rows_wmma<<<NBATCH * 256 / 8, 256, 0, stream>>>(Draw, Ttmp);
    conv_cols_wmma<<<NBATCH * 256 / 8, 256, 0, stream>>>(Ttmp, Smooth);

    overflow_kernel<<<nD / 256, 256, 0, stream>>>(Smooth, acc + NBATCH);

    finalize_kernel<<<1, 32, 0, stream>>>(acc, out);
}